// TropicalAffine_17411797418543
// MI455X (gfx1250) — compile-verified
//
#include <hip/hip_runtime.h>
#include <stdint.h>

// Problem constants (from reference)
#define B_N 4096
#define F_N 1024
#define LN_EPS 1e-5f

// Tropical matmul tiling: 64x128 output tile per 256-thread block -> 512 blocks
#define TM 64
#define TN 128
#define TK 16
#define NTHREADS 256

typedef __attribute__((ext_vector_type(2))) float f32x2;
typedef __attribute__((ext_vector_type(4))) float f32x4;
typedef __attribute__((ext_vector_type(8))) float f32x8;
typedef __attribute__((ext_vector_type(4))) unsigned int u32x4_t;
typedef __attribute__((ext_vector_type(8))) int i32x8_t;
typedef __attribute__((ext_vector_type(4))) int i32x4_t;

static __device__ __forceinline__ uint32_t lds_addr_of(const void* p) {
  return (uint32_t)(uintptr_t)p;  // low 32 bits of flat LDS address = LDS offset
}

// TDM 2D tile load: tile_d1 rows of tile_d0 contiguous fp32, row stride
// stride_elems, global -> LDS (packed). D# per CDNA5 ISA sec 8.3/8.4.
static __device__ __forceinline__ void tdm_load_tile_2d(
    uint32_t lds_byte_addr, const float* gptr,
    uint32_t tile_d0, uint32_t tile_d1, uint32_t stride_elems) {
  const uint64_t ga = (uint64_t)(uintptr_t)gptr;
  const uint32_t td0 = 1u << 24;
  const uint32_t td1 = 1u << 24;

  u32x4_t g0;
  g0[0] = 1u;                                                // count=1, user mode
  g0[1] = lds_byte_addr;                                     // lds_addr
  g0[2] = (uint32_t)ga;                                      // global_addr[31:0]
  g0[3] = ((uint32_t)(ga >> 32) & 0x01FFFFFFu) | (2u << 30); // addr[56:32]|type=2

  i32x8_t g1;
  g1[0] = (int)(2u << 16);                                   // data_size=2 (4B)
  g1[1] = (int)((td0 & 0xFFFFu) << 16);
  g1[2] = (int)((td0 >> 16) | ((td1 & 0xFFFFu) << 16));
  g1[3] = (int)((td1 >> 16) | ((tile_d0 & 0xFFFFu) << 16));
  g1[4] = (int)(tile_d1 & 0xFFFFu);
  g1[5] = (int)stride_elems;
  g1[6] = 0;
  g1[7] = 0;

  i32x4_t z4 = {0, 0, 0, 0};
#if defined(__clang_major__) && (__clang_major__ >= 23)
  i32x8_t z8 = {0, 0, 0, 0, 0, 0, 0, 0};
  __builtin_amdgcn_tensor_load_to_lds(g0, g1, z4, z4, z8, 0);
#else
  __builtin_amdgcn_tensor_load_to_lds(g0, g1, z4, z4, 0);
#endif
}

// ---------------------------------------------------------------------------
// Tropical max-plus matmul: out[r,c] = max(b[c], max_k xn[r,k] + W[k,c]).
// 64x128 tile/block, TDM double-buffered LDS, 4x8 register acc per thread.
// ---------------------------------------------------------------------------
#define UPD_ROW(i, xval, w0, w1)                          \
  acc[i][0] = fmaxf(acc[i][0], (xval) + (w0).x);          \
  acc[i][1] = fmaxf(acc[i][1], (xval) + (w0).y);          \
  acc[i][2] = fmaxf(acc[i][2], (xval) + (w0).z);          \
  acc[i][3] = fmaxf(acc[i][3], (xval) + (w0).w);          \
  acc[i][4] = fmaxf(acc[i][4], (xval) + (w1).x);          \
  acc[i][5] = fmaxf(acc[i][5], (xval) + (w1).y);          \
  acc[i][6] = fmaxf(acc[i][6], (xval) + (w1).z);          \
  acc[i][7] = fmaxf(acc[i][7], (xval) + (w1).w);

__global__ __launch_bounds__(NTHREADS) void tropical_mm_kernel(
    const float* __restrict__ xn, const float* __restrict__ W,
    const float* __restrict__ bias, float* __restrict__ out) {
  __shared__ __align__(16) float sX[2][TM * TK];  // [row][k], 4 KB per buffer
  __shared__ __align__(16) float sW[2][TK * TN];  // [k][col], 8 KB per buffer

  const int tid = threadIdx.x;
  const int tx = tid & 15;
  const int ty = tid >> 4;
  const int brow = blockIdx.y * TM;
  const int bcol = blockIdx.x * TN;

  float acc[4][8];
#pragma unroll
  for (int i = 0; i < 4; ++i)
#pragma unroll
    for (int j = 0; j < 8; ++j) acc[i][j] = -INFINITY;

  const bool issuer = (tid < 32);  // wave 0 drives the TDM
  const uint32_t ldsX[2] = {lds_addr_of(&sX[0][0]), lds_addr_of(&sX[1][0])};
  const uint32_t ldsW[2] = {lds_addr_of(&sW[0][0]), lds_addr_of(&sW[1][0])};

  if (issuer) {
    tdm_load_tile_2d(ldsX[0], xn + (size_t)brow * F_N, TK, TM, F_N);
    tdm_load_tile_2d(ldsW[0], W + (size_t)bcol, TN, TK, F_N);
  }

  const int KSTEPS = F_N / TK;  // 64
  for (int kt = 0; kt < KSTEPS; ++kt) {
    const int cur = kt & 1;
    const int nxt = cur ^ 1;
    if (issuer) {
      if (kt + 1 < KSTEPS) {
        const int k0 = (kt + 1) * TK;
        tdm_load_tile_2d(ldsX[nxt], xn + (size_t)brow * F_N + k0, TK, TM, F_N);
        tdm_load_tile_2d(ldsW[nxt], W + (size_t)k0 * F_N + bcol, TN, TK, F_N);
        __builtin_amdgcn_s_wait_tensorcnt(2);  // stage kt done (in-order TDM)
      } else {
        __builtin_amdgcn_s_wait_tensorcnt(0);
      }
    }
    __syncthreads();

    const float* xs = &sX[cur][0];
    const float* ws = &sW[cur][0];
#pragma unroll
    for (int k4 = 0; k4 < TK; k4 += 4) {
      f32x4 wv[4][2];
#pragma unroll
      for (int kk = 0; kk < 4; ++kk) {
        wv[kk][0] = *(const f32x4*)(ws + (k4 + kk) * TN + tx * 8);
        wv[kk][1] = *(const f32x4*)(ws + (k4 + kk) * TN + tx * 8 + 4);
      }
#pragma unroll
      for (int i = 0; i < 4; ++i) {
        f32x4 xv = *(const f32x4*)(xs + (ty * 4 + i) * TK + k4);
        UPD_ROW(i, xv.x, wv[0][0], wv[0][1]);
        UPD_ROW(i, xv.y, wv[1][0], wv[1][1]);
        UPD_ROW(i, xv.z, wv[2][0], wv[2][1]);
        UPD_ROW(i, xv.w, wv[3][0], wv[3][1]);
      }
    }
    __syncthreads();
  }

  f32x4 b0 = *(const f32x4*)(bias + bcol + tx * 8);
  f32x4 b1 = *(const f32x4*)(bias + bcol + tx * 8 + 4);
#pragma unroll
  for (int i = 0; i < 4; ++i) {
    const size_t r = (size_t)(brow + ty * 4 + i);
    f32x4 o0, o1;
    o0.x = fmaxf(acc[i][0], b0.x);
    o0.y = fmaxf(acc[i][1], b0.y);
    o0.z = fmaxf(acc[i][2], b0.z);
    o0.w = fmaxf(acc[i][3], b0.w);
    o1.x = fmaxf(acc[i][4], b1.x);
    o1.y = fmaxf(acc[i][5], b1.y);
    o1.z = fmaxf(acc[i][6], b1.z);
    o1.w = fmaxf(acc[i][7], b1.w);
    __builtin_nontemporal_store(o0, (f32x4*)(out + r * F_N + bcol + tx * 8));
    __builtin_nontemporal_store(o1, (f32x4*)(out + r * F_N + bcol + tx * 8 + 4));
  }
}

// ---------------------------------------------------------------------------
// LayerNorm stats via the matrix pipe: row sums are X*ones and (X.X)*ones,
// i.e. chained V_WMMA_F32_16X16X4_F32 accumulations (full fp32 precision).
// One wave owns 16 rows. A-fragment layout per ISA 7.12.2:
//   lane = 16*h + m  ->  A holds x[row m][k0+2h], x[row m][k0+2h+1]
// B = ones (layout-invariant). D: lane L, vgpr j = rowsum(j + 8*(L>=16)).
// Writes {mu, rstd} per row. Two independent WMMA chains (sum / sumsq)
// interleave for XDL pipelining. Kernel is HBM-bound either way.
// ---------------------------------------------------------------------------
#if __has_builtin(__builtin_amdgcn_wmma_f32_16x16x4_f32)
#define HAVE_WMMA_F32X4 1
#else
#define HAVE_WMMA_F32X4 0
#endif

__global__ __launch_bounds__(NTHREADS) void ln_stats_kernel(
    const float* __restrict__ x, float* __restrict__ stats) {
  const int wave = threadIdx.x >> 5;
  const int lane = threadIdx.x & 31;
  const int half = lane >> 4;
  const int m = lane & 15;
  const int row0 = blockIdx.x * (NTHREADS / 32) * 16 + wave * 16;

#if HAVE_WMMA_F32X4
  const float* xr = x + (size_t)(row0 + m) * F_N + 2 * half;
  f32x8 cs = {0, 0, 0, 0, 0, 0, 0, 0};
  f32x8 cq = {0, 0, 0, 0, 0, 0, 0, 0};
  const f32x2 ones = {1.0f, 1.0f};
#pragma unroll 4
  for (int k0 = 0; k0 < F_N; k0 += 4) {
    f32x2 a = *(const f32x2*)(xr + k0);
    f32x2 aq = a * a;
    cs = __builtin_amdgcn_wmma_f32_16x16x4_f32(false, a, false, ones,
                                               (short)0, cs, false, false);
    cq = __builtin_amdgcn_wmma_f32_16x16x4_f32(false, aq, false, ones,
                                               (short)0, cq, false, false);
  }
  if ((lane & 15) == 0) {  // lanes 0 and 16 hold rows 0..7 / 8..15 in vgprs 0..7
    const int base = row0 + half * 8;
#pragma unroll
    for (int j = 0; j < 8; ++j) {
      const float mu = cs[j] * (1.0f / F_N);
      const float var = cq[j] * (1.0f / F_N) - mu * mu;
      f32x2 st = {mu, rsqrtf(var + LN_EPS)};
      *(f32x2*)(stats + 2 * (size_t)(base + j)) = st;
    }
  }
#else
  // Fallback: lane m reduces row m with VALU (halves 0/1 combine via shfl).
  const float* xr = x + (size_t)(row0 + m) * F_N;
  float s = 0.0f, q = 0.0f;
  for (int k = half * 512; k < half * 512 + 512; k += 4) {
    f32x4 v = *(const f32x4*)(xr + k);
    s += v.x + v.y + v.z + v.w;
    q += v.x * v.x + v.y * v.y + v.z * v.z + v.w * v.w;
  }
  s += __shfl_xor(s, 16, 32);
  q += __shfl_xor(q, 16, 32);
  if (half == 0) {
    const float mu = s * (1.0f / F_N);
    const float var = q * (1.0f / F_N) - mu * mu;
    f32x2 st = {mu, rsqrtf(var + LN_EPS)};
    *(f32x2*)(stats + 2 * (size_t)(row0 + m)) = st;
  }
#endif
}

// Coalesced LayerNorm apply: xn[r,c] = (x[r,c]-mu[r])*rstd[r]*gamma[c]+beta[c].
__global__ __launch_bounds__(NTHREADS) void ln_apply_kernel(
    const float* __restrict__ x, const float* __restrict__ gamma,
    const float* __restrict__ beta, const float* __restrict__ stats,
    float* __restrict__ xn) {
  const int tid = threadIdx.x;
  const int row = blockIdx.x;
  const float mu = stats[2 * (size_t)row];
  const float rstd = stats[2 * (size_t)row + 1];

  f32x4 v = __builtin_nontemporal_load(
      (const f32x4*)(x + (size_t)row * F_N + tid * 4));
  f32x4 g = *(const f32x4*)(gamma + tid * 4);
  f32x4 be = *(const f32x4*)(beta + tid * 4);
  f32x4 o;
  o.x = (v.x - mu) * rstd * g.x + be.x;
  o.y = (v.y - mu) * rstd * g.y + be.y;
  o.z = (v.z - mu) * rstd * g.z + be.z;
  o.w = (v.w - mu) * rstd * g.w + be.w;
  *(f32x4*)(xn + (size_t)row * F_N + tid * 4) = o;  // stays hot in 192MB L2
}

extern "C" void kernel_launch(void* const* d_in, const int* in_sizes, int n_in,
                              void* d_out, int out_size, void* d_ws, size_t ws_size,
                              hipStream_t stream) {
  const float* x     = (const float*)d_in[0];
  const float* W     = (const float*)d_in[1];
  const float* b     = (const float*)d_in[2];
  const float* gamma = (const float*)d_in[3];
  const float* beta  = (const float*)d_in[4];
  float* out = (float*)d_out;
  float* xn = (float*)d_ws;                            // 16 MB
  float* stats = (float*)((char*)d_ws + (size_t)B_N * F_N * 4);  // 32 KB

  // 8 waves/block, 16 rows/wave -> 128 rows/block -> 32 blocks
  ln_stats_kernel<<<B_N / 128, NTHREADS, 0, stream>>>(x, stats);
  ln_apply_kernel<<<B_N, NTHREADS, 0, stream>>>(x, gamma, beta, stats, xn);

  dim3 grid(F_N / TN, B_N / TM);  // (8, 64) = 512 blocks
  tropical_mm_kernel<<<grid, NTHREADS, 0, stream>>>(xn, W, b, out);
}